// QuanvLayer_7017976562077
// MI455X (gfx1250) — compile-verified
//
#include <hip/hip_runtime.h>
#include <math.h>

typedef float v2f __attribute__((ext_vector_type(2)));
typedef float v8f __attribute__((ext_vector_type(8)));

// ---------------------------------------------------------------------------
// Kernel A: build the fixed circuit unitary U (16x16 complex) from the 7
// ansatz params. Stored ROW-MAJOR:  Are[n][k] = Re U[n][k], Aim likewise,
// so it can be loaded directly as the WMMA A-operand for  Psi^T = U * E^T.
// One wave; lane j evolves basis column j. Qubit q <-> bit (3-q) of the index.
// ---------------------------------------------------------------------------

#define APPLY1Q(MASK, U00R,U00I,U01R,U01I,U10R,U10I,U11R,U11I)                 \
  do { _Pragma("unroll")                                                       \
    for (int s_ = 0; s_ < 16; ++s_) {                                          \
      if ((s_ & (MASK)) == 0) {                                                \
        const int s1_ = s_ | (MASK);                                           \
        const float ar_ = vr[s_], ai_ = vi[s_];                                \
        const float br_ = vr[s1_], bi_ = vi[s1_];                              \
        vr[s_]  = (U00R)*ar_ - (U00I)*ai_ + (U01R)*br_ - (U01I)*bi_;           \
        vi[s_]  = (U00R)*ai_ + (U00I)*ar_ + (U01R)*bi_ + (U01I)*br_;           \
        vr[s1_] = (U10R)*ar_ - (U10I)*ai_ + (U11R)*br_ - (U11I)*bi_;           \
        vi[s1_] = (U10R)*ai_ + (U10I)*ar_ + (U11R)*bi_ + (U11I)*br_;           \
      } } } while (0)

#define APPLY_CNOT(CMASK, TMASK)                                               \
  do { _Pragma("unroll")                                                       \
    for (int s_ = 0; s_ < 16; ++s_) {                                          \
      if ((s_ & (CMASK)) && !(s_ & (TMASK))) {                                 \
        const int s1_ = s_ ^ (TMASK);                                          \
        float tr_ = vr[s_], ti_ = vi[s_];                                      \
        vr[s_] = vr[s1_]; vi[s_] = vi[s1_];                                    \
        vr[s1_] = tr_;    vi[s1_] = ti_;                                       \
      } } } while (0)

__global__ void quanv_build_unitary(const float* __restrict__ params,
                                    float* __restrict__ are,
                                    float* __restrict__ aim) {
    const int j = threadIdx.x;
    if (j >= 16) return;

    float vr[16], vi[16];
    #pragma unroll
    for (int i = 0; i < 16; ++i) { vr[i] = (i == j) ? 1.f : 0.f; vi[i] = 0.f; }

    float c, s;
    // layer 1: rx(p0) q0(mask8), ry(p1) q1(mask4), rz(p2) q2(mask2), H q3(mask1)
    c = cosf(0.5f * params[0]); s = sinf(0.5f * params[0]);
    APPLY1Q(8,  c, 0.f,  0.f, -s,   0.f, -s,   c, 0.f);          // RX
    c = cosf(0.5f * params[1]); s = sinf(0.5f * params[1]);
    APPLY1Q(4,  c, 0.f,  -s, 0.f,   s, 0.f,    c, 0.f);          // RY
    c = cosf(0.5f * params[2]); s = sinf(0.5f * params[2]);
    APPLY1Q(2,  c, -s,   0.f, 0.f,  0.f, 0.f,  c,  s);           // RZ
    {
        const float r2 = 0.70710678118654752440f;
        APPLY1Q(1,  r2, 0.f,  r2, 0.f,  r2, 0.f,  -r2, 0.f);     // H
    }
    // layer 2: CNOT(0,1) (1,2) (2,3) (3,0)
    APPLY_CNOT(8, 4);
    APPLY_CNOT(4, 2);
    APPLY_CNOT(2, 1);
    APPLY_CNOT(1, 8);
    // layer 3: ry(p3..p6) on q0..q3
    c = cosf(0.5f * params[3]); s = sinf(0.5f * params[3]);
    APPLY1Q(8,  c, 0.f, -s, 0.f,  s, 0.f,  c, 0.f);
    c = cosf(0.5f * params[4]); s = sinf(0.5f * params[4]);
    APPLY1Q(4,  c, 0.f, -s, 0.f,  s, 0.f,  c, 0.f);
    c = cosf(0.5f * params[5]); s = sinf(0.5f * params[5]);
    APPLY1Q(2,  c, 0.f, -s, 0.f,  s, 0.f,  c, 0.f);
    c = cosf(0.5f * params[6]); s = sinf(0.5f * params[6]);
    APPLY1Q(1,  c, 0.f, -s, 0.f,  s, 0.f,  c, 0.f);

    // column j of U == v  ->  row-major A: A[n][k=j] = U[n][j] = v[n]
    #pragma unroll
    for (int n = 0; n < 16; ++n) {
        are[n * 16 + j] = vr[n];
        aim[n * 16 + j] = vi[n];
    }
}

// ---------------------------------------------------------------------------
// Kernel B: per wave, tiles of 16 patches.
//   Psi^T[n][patch] = sum_k U[n][k] * E^T[k][patch]
//   A = U (loop-invariant A fragments, re+im), B = E^T built per tile.
//   D layout: register r holds state n = r (lanes 0-15) / r+8 (lanes 16-31),
//   column = patch = lane&15  ->  all 16 states of a patch live in one lane
//   pair's 8 registers. <Z_q> = signed register sums (+ one shfl_xor(16) for
//   the half-split qubit 0). Coalesced float4 store per patch.
// ---------------------------------------------------------------------------
__global__ void quanv_main(const float* __restrict__ x,
                           const float* __restrict__ are_g,
                           const float* __restrict__ aim_g,
                           float* __restrict__ out,
                           int npatch, int ntiles) {
    const int lane  = threadIdx.x & 31;
    const int wib   = threadIdx.x >> 5;
    const int wpb   = blockDim.x >> 5;
    const int gwave = blockIdx.x * wpb + wib;
    const int nwave = gridDim.x * wpb;

    const int lb    = lane & 15;
    const int hi    = lane >> 4;          // 0: lanes 0-15, 1: lanes 16-31
    const int base2 = hi << 1;            // K offset within a 4-chunk

    // Loop-invariant A fragments: A[m=lb][k], K split across wave halves.
    v2f aRe[4], aIm[4];
    #pragma unroll
    for (int j = 0; j < 4; ++j) {
        const int k0 = 4 * j + base2;
        aRe[j].x = are_g[lb * 16 + k0    ];
        aRe[j].y = are_g[lb * 16 + k0 + 1];
        aIm[j].x = aim_g[lb * 16 + k0    ];
        aIm[j].y = aim_g[lb * 16 + k0 + 1];
    }

    const float HPI = 1.57079632679489662f; // encoder RY(pi*x): gate angle/2

    for (int tile = gwave; tile < ntiles; tile += nwave) {
        int m = tile * 16 + lb;
        if (m >= npatch) m = npatch - 1;             // clamp (N is 16-divisible)
        const float4 ang = ((const float4*)x)[m];

        const float a0 = HPI * ang.x, a1 = HPI * ang.y;
        const float a2 = HPI * ang.z, a3 = HPI * ang.w;
        const float c0 = __cosf(a0), s0 = __sinf(a0);
        const float c1 = __cosf(a1), s1 = __sinf(a1);
        const float c2 = __cosf(a2), s2 = __sinf(a2);
        const float c3 = __cosf(a3), s3 = __sinf(a3);

        // E(k) for this lane's patch lb: k = 4j + base2 + {0,1}
        const float h = hi ? s2 : c2;       // bit1 of k fixed by wave half
        const float u = h * c3;             // bit0 = 0
        const float w = h * s3;             // bit0 = 1
        float q01[4];
        q01[0] = c0 * c1; q01[1] = c0 * s1; q01[2] = s0 * c1; q01[3] = s0 * s1;

        v8f cre = {}; v8f cim = {};
        #pragma unroll
        for (int j = 0; j < 4; ++j) {
            v2f e;                          // B fragment rows k=4j+base2(+1)
            e.x = q01[j] * u;
            e.y = q01[j] * w;
            cre = __builtin_amdgcn_wmma_f32_16x16x4_f32(
                      false, aRe[j], false, e, (short)0, cre, false, false);
            cim = __builtin_amdgcn_wmma_f32_16x16x4_f32(
                      false, aIm[j], false, e, (short)0, cim, false, false);
        }

        // probabilities: p[r] = |psi_{n=r+8*hi}|^2 for patch lb
        v8f p;
        #pragma unroll
        for (int r = 0; r < 8; ++r) p[r] = cre[r] * cre[r] + cim[r] * cim[r];

        // in-register signed sums over r (bits of the state index n = r+8*hi)
        const float e0 = p[0] + p[1], d0 = p[0] - p[1];
        const float e1 = p[2] + p[3], d1 = p[2] - p[3];
        const float e2 = p[4] + p[5], d2 = p[4] - p[5];
        const float e3 = p[6] + p[7], d3 = p[6] - p[7];
        const float u0 = e0 + e1, u1 = e2 + e3;
        const float tot = u0 + u1;                       // sum over this half
        const float t1  = u0 - u1;                       // sign by bit2 -> q=1
        const float t2  = (e0 - e1) + (e2 - e3);         // sign by bit1 -> q=2
        const float t3  = (d0 + d1) + (d2 + d3);         // sign by bit0 -> q=3

        // combine wave halves (bit3 of n -> q=0 gets a sign flip on hi half)
        const float oT = __shfl_xor(tot, 16, 32);
        const float o1 = __shfl_xor(t1,  16, 32);
        const float o2 = __shfl_xor(t2,  16, 32);
        const float o3 = __shfl_xor(t3,  16, 32);
        const float z0 = hi ? (oT - tot) : (tot - oT);
        const float z1 = t1 + o1;
        const float z2 = t2 + o2;
        const float z3 = t3 + o3;

        if (!hi) {
            const int patch = tile * 16 + lb;
            if (patch < npatch) {
                float4 o4; o4.x = z0; o4.y = z1; o4.z = z2; o4.w = z3;
                *(float4*)(out + patch * 4) = o4;        // 16B-aligned, coalesced
            }
        }
    }
}

extern "C" void kernel_launch(void* const* d_in, const int* in_sizes, int n_in,
                              void* d_out, int out_size, void* d_ws, size_t ws_size,
                              hipStream_t stream) {
    const float* x      = (const float*)d_in[0];
    const float* params = (const float*)d_in[1];
    float*       out    = (float*)d_out;

    float* are = (float*)d_ws;        // 256 floats
    float* aim = are + 256;           // 256 floats (ws >= 2 KB)

    const int npatch = in_sizes[0] / 4;
    const int ntiles = (npatch + 15) / 16;

    quanv_build_unitary<<<1, 32, 0, stream>>>(params, are, aim);

    const int threads       = 256;            // 8 waves/block
    const int wavesPerBlock = threads / 32;
    const int tilesPerWave  = 8;
    int blocks = (ntiles + wavesPerBlock * tilesPerWave - 1) /
                 (wavesPerBlock * tilesPerWave);
    if (blocks < 1) blocks = 1;

    quanv_main<<<blocks, threads, 0, stream>>>(x, are, aim, out, npatch, ntiles);
}